// QDDPM_cpu_66571993089069
// MI455X (gfx1250) — compile-verified
//
#include <hip/hip_runtime.h>

// QDDPM backward-circuit kernel for MI455X (gfx1250).
//
// Strategy: one workgroup (256 threads = 8 wave32) per batch element.
// The full 8192-complex state lives in LDS (64 KB of CDNA5's 320 KB WGP LDS).
// Per layer, the 13 commuting 1-qubit gates are fused:
//   - qubits {0-3},{4-7},{8-11}: Kronecker-fused into 16x16 complex matrices,
//     applied as in-place 16x16x512 complex GEMMs with V_WMMA_F32_16X16X4_F32
//     (fp32 matrix core path; 4 chained K=4 chunks per 16-col tile, separate
//     real/imag accumulators; 16 WMMAs per tile).
//   - qubit 12 (right=1) + the brickwork-CZ diagonal sign (= parity of
//     popc(idx & idx>>1) over 13 bits) in one scalar pass.
// The initial state load uses GLOBAL_LOAD_ASYNC_TO_LDS_B128 (ASYNCcnt path)
// when the toolchain exposes it, overlapped with layer-0 gate construction;
// drained with s_wait_asynccnt 0 just before the first state-consuming barrier.
// After 6 layers: ancilla post-selection, block norm reduction, normalized
// interleaved (re,im) store.
//
// HBM traffic: 32 MB in + 16 MB out total (vs ~5 GB for gate-at-a-time),
// so ~2 us of DRAM time at 23.3 TB/s; everything else runs out of LDS.

typedef float v2f __attribute__((ext_vector_type(2)));
typedef float v8f __attribute__((ext_vector_type(8)));

#if defined(__has_builtin)
#if __has_builtin(__builtin_amdgcn_global_load_async_to_lds_b128)
#define QDDPM_ASYNC_LDS 1
#endif
#endif

#ifdef QDDPM_ASYNC_LDS
// ROCm clang declares the async-LDS builtins with typed vector pointers:
//   (v4i addrspace(1)*, v4i addrspace(3)*, imm offset, imm cpol)
typedef int v4i_vs __attribute__((vector_size(16)));
typedef __attribute__((address_space(1))) v4i_vs* gptr_b128;
typedef __attribute__((address_space(3))) v4i_vs* lptr_b128;
#endif

static __device__ __forceinline__ void wait_async0() {
#ifdef QDDPM_ASYNC_LDS
#if __has_builtin(__builtin_amdgcn_s_wait_asynccnt)
  __builtin_amdgcn_s_wait_asynccnt(0);
#else
  asm volatile("s_wait_asynccnt 0" ::: "memory");
#endif
#endif
}

static __device__ __forceinline__ v8f wmma_k4(v2f a, v2f b, v8f c) {
  // D(16x16,f32) = A(16x4,f32) * B(4x16,f32) + C
  return __builtin_amdgcn_wmma_f32_16x16x4_f32(
      /*neg_a=*/false, a, /*neg_b=*/false, b,
      /*c_mod=*/(short)0, c, /*reuse_a=*/false, /*reuse_b=*/false);
}

#define NQ     13
#define DIMT   8192
#define DD     4096
#define LAYERS 6

__global__ __launch_bounds__(256, 1)
void qddpm_backward_kernel(const float* __restrict__ in_re,
                           const float* __restrict__ in_im,
                           const float* __restrict__ params,
                           const int*   __restrict__ m_res,
                           float* __restrict__ out)
{
  __shared__ float sre[DIMT];        // 32 KB
  __shared__ float sim[DIMT];        // 32 KB
  __shared__ float gq[NQ][8];        // per-qubit fused RY*RX entries {a,b,c,d} complex
  __shared__ float Gre[3][16][16];   // group matrices (real)
  __shared__ float Gim[3][16][16];   // group matrices (imag)

  const int t    = threadIdx.x;
  const int b    = blockIdx.x;
  const int lane = t & 31;
  const int wave = t >> 5;
  const int half = lane >> 4;   // 0: lanes 0-15, 1: lanes 16-31
  const int l15  = lane & 15;

  // ---- bring state into LDS ----
#ifdef QDDPM_ASYNC_LDS
  {
    // Direct HBM->LDS async copy (B128 per lane), overlapped with the
    // layer-0 gate construction below; drained before the first GEMM.
    const float* gr = in_re + (size_t)b * DIMT;
    const float* gi = in_im + (size_t)b * DIMT;
#pragma unroll
    for (int k = 0; k < 8; ++k) {
      const int idx = (t + k * 256) * 4;   // float index, 16B granules
      __builtin_amdgcn_global_load_async_to_lds_b128(
          (gptr_b128)(gr + idx), (lptr_b128)(sre + idx), 0, 0);
      __builtin_amdgcn_global_load_async_to_lds_b128(
          (gptr_b128)(gi + idx), (lptr_b128)(sim + idx), 0, 0);
    }
  }
#else
  {
    const float4* gr = (const float4*)(in_re + (size_t)b * DIMT);
    const float4* gi = (const float4*)(in_im + (size_t)b * DIMT);
    float4* lr = (float4*)sre;
    float4* li = (float4*)sim;
#pragma unroll
    for (int k = 0; k < 8; ++k) {
      lr[t + k * 256] = gr[t + k * 256];
      li[t + k * 256] = gi[t + k * 256];
    }
  }
  __syncthreads();
#endif

  for (int l = 0; l < LAYERS; ++l) {
    // ---- per-qubit fused gate U = RY*RX:  a=cycx+i sysx, b=-sycx-i cysx,
    //      c=sycx-i cysx, d=cycx-i sysx  (independent of the state) ----
    if (t < NQ) {
      float tx = params[2 * NQ * l + t];
      float ty = params[2 * NQ * l + NQ + t];
      float sx, cx, sy, cy;
      __sincosf(0.5f * tx, &sx, &cx);
      __sincosf(0.5f * ty, &sy, &cy);
      gq[t][0] =  cy * cx;  gq[t][1] =  sy * sx;   // a
      gq[t][2] = -sy * cx;  gq[t][3] = -cy * sx;   // b
      gq[t][4] =  sy * cx;  gq[t][5] = -cy * sx;   // c
      gq[t][6] =  cy * cx;  gq[t][7] = -sy * sx;   // d
    }
    __syncthreads();

    // ---- build 16x16 Kronecker matrices for qubit groups {0-3},{4-7},{8-11} ----
    {
      const int i = t >> 4, j = t & 15;
#pragma unroll
      for (int g = 0; g < 3; ++g) {
        float pr = 1.f, pi = 0.f;
#pragma unroll
        for (int k = 0; k < 4; ++k) {
          const int q   = 4 * g + k;
          const int sel = ((((i >> (3 - k)) & 1) << 1) | ((j >> (3 - k)) & 1)) << 1;
          const float ur = gq[q][sel], ui = gq[q][sel + 1];
          const float nr = pr * ur - pi * ui;
          pi = pr * ui + pi * ur;
          pr = nr;
        }
        Gre[g][i][j] = pr;
        Gim[g][i][j] = pi;
      }
    }
    if (l == 0) wait_async0();   // state must be resident before first GEMM
    __syncthreads();

    // ---- apply the three 16x16 complex matrices with f32 WMMA, in place ----
    for (int g = 0; g < 3; ++g) {
      const int rg = 512 >> (4 * g);  // stride of the 4-bit group field: 512, 32, 2

      // A fragments (constant across this group's tiles):
      // lanes 0-15: M=l15, VGPR0/1 = K = 4kc+0 / 4kc+1; lanes 16-31: K = 4kc+2 / 4kc+3
      v2f aRe[4], aIm[4];
#pragma unroll
      for (int kc = 0; kc < 4; ++kc) {
        const int kk = 4 * kc + 2 * half;
        v2f ar = { Gre[g][l15][kk], Gre[g][l15][kk + 1] };
        v2f ai = { Gim[g][l15][kk], Gim[g][l15][kk + 1] };
        aRe[kc] = ar;
        aIm[kc] = ai;
      }

#pragma unroll 1
      for (int tt = 0; tt < 4; ++tt) {          // 4 tiles per wave, 32 tiles total
        const int col  = (wave * 4 + tt) * 16 + l15;
        // element index for (row j, column col): high bits of col shifted past
        // the 4-bit group field, plus j*rg, plus low bits of col
        const int base = ((col & ~(rg - 1)) << 4) | (col & (rg - 1));

        v2f bRe[4], bIm[4];
#pragma unroll
        for (int kc = 0; kc < 4; ++kc) {
          const int j0 = 4 * kc + 2 * half;
          v2f br = { sre[base + j0 * rg], sre[base + (j0 + 1) * rg] };
          v2f bi = { sim[base + j0 * rg], sim[base + (j0 + 1) * rg] };
          bRe[kc] = br;
          bIm[kc] = bi;
        }

        v8f dre = {};
        v8f dim = {};
#pragma unroll
        for (int kc = 0; kc < 4; ++kc) dre = wmma_k4(aRe[kc], bRe[kc], dre);
#pragma unroll
        for (int kc = 0; kc < 4; ++kc) { v2f na = -aIm[kc]; dre = wmma_k4(na, bIm[kc], dre); }
#pragma unroll
        for (int kc = 0; kc < 4; ++kc) dim = wmma_k4(aRe[kc], bIm[kc], dim);
#pragma unroll
        for (int kc = 0; kc < 4; ++kc) dim = wmma_k4(aIm[kc], bRe[kc], dim);

        // D layout: VGPR v -> (M = v + 8*half, N = l15); columns of this tile
        // are owned exclusively by this wave -> safe in-place writeback.
#pragma unroll
        for (int v = 0; v < 8; ++v) {
          const int e = base + (v + 8 * half) * rg;
          sre[e] = dre[v];
          sim[e] = dim[v];
        }
      }
      __syncthreads();
    }

    // ---- qubit 12 (LSB) 2x2 gate + brickwork-CZ diagonal sign ----
    {
      const float ar = gq[12][0], ai = gq[12][1], br = gq[12][2], bi = gq[12][3];
      const float cr = gq[12][4], ci = gq[12][5], er = gq[12][6], ei = gq[12][7];
#pragma unroll
      for (int k = 0; k < 16; ++k) {
        const int e0 = 2 * (t + k * 256);
        const int e1 = e0 + 1;
        const float s0r = sre[e0], s0i = sim[e0];
        const float s1r = sre[e1], s1i = sim[e1];
        const float n0r = ar * s0r - ai * s0i + br * s1r - bi * s1i;
        const float n0i = ar * s0i + ai * s0r + br * s1i + bi * s1r;
        const float n1r = cr * s0r - ci * s0i + er * s1r - ei * s1i;
        const float n1i = cr * s0i + ci * s0r + er * s1i + ei * s1r;
        const float sg0 = (__popc(e0 & (e0 >> 1)) & 1) ? -1.f : 1.f;
        const float sg1 = (__popc(e1 & (e1 >> 1)) & 1) ? -1.f : 1.f;
        sre[e0] = n0r * sg0;  sim[e0] = n0i * sg0;
        sre[e1] = n1r * sg1;  sim[e1] = n1i * sg1;
      }
    }
    __syncthreads();
  }

  // ---- post-select ancilla (MSB), normalize, write [4096][2] f32 ----
  const int off = (m_res[b] == 0) ? 0 : DD;
  float part = 0.f;
#pragma unroll
  for (int k = 0; k < 16; ++k) {
    const int e = off + t + k * 256;
    part += sre[e] * sre[e] + sim[e] * sim[e];
  }
  float* red = &Gre[0][0][0];   // reuse group-matrix LDS as reduction scratch
  red[t] = part;
  __syncthreads();
  for (int s = 128; s > 0; s >>= 1) {
    if (t < s) red[t] += red[t + s];
    __syncthreads();
  }
  const float inv = rsqrtf(red[0]);

  float2* o2 = (float2*)out + (size_t)b * DD;
#pragma unroll
  for (int k = 0; k < 16; ++k) {
    const int e = t + k * 256;
    float2 v;
    v.x = sre[off + e] * inv;
    v.y = sim[off + e] * inv;
    o2[e] = v;
  }
}

extern "C" void kernel_launch(void* const* d_in, const int* in_sizes, int n_in,
                              void* d_out, int out_size, void* d_ws, size_t ws_size,
                              hipStream_t stream) {
  (void)n_in; (void)out_size; (void)d_ws; (void)ws_size;
  const float* in_re  = (const float*)d_in[0];
  const float* in_im  = (const float*)d_in[1];
  const float* params = (const float*)d_in[2];
  const int*   m_res  = (const int*)d_in[3];
  float*       outp   = (float*)d_out;

  const int B = in_sizes[3];  // 512 batch elements, one workgroup each
  qddpm_backward_kernel<<<dim3(B), dim3(256), 0, stream>>>(in_re, in_im, params, m_res, outp);
}